// TopK_63960652972425
// MI455X (gfx1250) — compile-verified
//
#include <hip/hip_runtime.h>
#include <stdint.h>

#define TK_ROWS 8192
#define TK_COLS 16384
#define TK_K    64
#define TK_NT   256
#define TK_CHUNKS (TK_COLS / (TK_NT * 4))   // 16 float4 chunks per thread

typedef float f32x4 __attribute__((ext_vector_type(4)));

// Monotonic float->u32 key: larger float => larger key.
__device__ __forceinline__ uint32_t tk_key(float f) {
    uint32_t u = __float_as_uint(f);
    return (u & 0x80000000u) ? ~u : (u | 0x80000000u);
}

__global__ __launch_bounds__(TK_NT) void TopK_63960652972425_kernel(
        const float* __restrict__ x, float* __restrict__ out) {
    __shared__ float    row[TK_COLS];     // 64 KB staged row
    __shared__ uint32_t hist[256];        // radix histogram
    __shared__ uint32_t scan[256];        // suffix-sum workspace
    __shared__ uint32_t s_prefix;
    __shared__ uint32_t s_kneed;
    __shared__ int      s_tie;

    const int    tid   = (int)threadIdx.x;
    const size_t rbase = (size_t)blockIdx.x * (size_t)TK_COLS;
    const float* src   = x + rbase;

    // ---- Stage the row into LDS with CDNA5 async copies (ASYNCcnt path) ----
    const uint32_t lds_base = (uint32_t)(uintptr_t)(&row[0]);
    #pragma unroll
    for (int i = 0; i < TK_CHUNKS; ++i) {
        const int      c        = tid + TK_NT * i;          // float4 chunk index
        const uint32_t lds_addr = lds_base + (uint32_t)(c * 16);
        const float*   g        = src + c * 4;
        asm volatile("global_load_async_to_lds_b128 %0, %1, off"
                     :: "v"(lds_addr), "v"(g)
                     : "memory");
    }
    asm volatile("s_wait_asynccnt 0x0" ::: "memory");
    __syncthreads();

    // ---- Radix-select the top-64 threshold key: 4 x 8-bit rounds ----
    uint32_t prefix = 0;       // known high bits of threshold key
    uint32_t kneed  = TK_K;    // elements still needed inside prefix class
    #pragma unroll 1
    for (int round = 0; round < 4; ++round) {
        const int shift = 24 - 8 * round;
        hist[tid] = 0;
        __syncthreads();

        // Accumulate 256-bin histogram of current byte within prefix class.
        const uint32_t hmask = (round == 0) ? 0u : (0xFFFFFFFFu << (shift + 8));
        #pragma unroll
        for (int i = 0; i < TK_CHUNKS; ++i) {
            const int c = tid + TK_NT * i;
            const f32x4 v = ((const f32x4*)row)[c];
            #pragma unroll
            for (int j = 0; j < 4; ++j) {
                const uint32_t key = tk_key(v[j]);
                if ((key & hmask) == prefix)
                    atomicAdd(&hist[(key >> shift) & 255u], 1u);
            }
        }
        __syncthreads();

        // Parallel suffix sum: scan[t] = sum_{b >= t} hist[b]   (Hillis-Steele)
        scan[tid] = hist[tid];
        __syncthreads();
        #pragma unroll
        for (int off = 1; off < 256; off <<= 1) {
            const uint32_t add = (tid + off < 256) ? scan[tid + off] : 0u;
            __syncthreads();
            scan[tid] += add;
            __syncthreads();
        }

        // Unique thread with suffix[t] >= kneed > suffix[t+1] owns the bin.
        const uint32_t suff     = scan[tid];
        const uint32_t suffNext = (tid < 255) ? scan[tid + 1] : 0u;
        if (suff >= kneed && suffNext < kneed) {
            s_prefix = prefix | ((uint32_t)tid << shift);
            s_kneed  = kneed - suffNext;
        }
        __syncthreads();
        prefix = s_prefix;
        kneed  = s_kneed;
    }

    const uint32_t T = prefix;           // exact key of the 64th-largest value
    if (tid == 0) s_tie = (int)kneed;    // tie budget at the threshold
    __syncthreads();

    // ---- Scatter: relu(top-64) into dense output, zeros elsewhere ----
    float* dst = out + rbase;
    #pragma unroll
    for (int i = 0; i < TK_CHUNKS; ++i) {
        const int c = tid + TK_NT * i;
        const f32x4 v = ((const f32x4*)row)[c];
        f32x4 o;
        #pragma unroll
        for (int j = 0; j < 4; ++j) {
            const float    val = v[j];
            const uint32_t key = tk_key(val);
            float res = 0.0f;
            if (key > T) {
                res = val > 0.0f ? val : 0.0f;
            } else if (key == T) {
                const int old = atomicSub(&s_tie, 1);
                if (old > 0) res = val > 0.0f ? val : 0.0f;
            }
            o[j] = res;
        }
        __builtin_nontemporal_store(o, (f32x4*)(dst + c * 4));
    }
}

extern "C" void kernel_launch(void* const* d_in, const int* in_sizes, int n_in,
                              void* d_out, int out_size, void* d_ws, size_t ws_size,
                              hipStream_t stream) {
    const float* x   = (const float*)d_in[0];
    float*       out = (float*)d_out;
    dim3 grid(TK_ROWS), block(TK_NT);
    TopK_63960652972425_kernel<<<grid, block, 0, stream>>>(x, out);
}